// Retina_17643725652195
// MI455X (gfx1250) — compile-verified
//
#include <hip/hip_runtime.h>

// Retina / foveated glimpse: per-image clamped 128x128 crop -> 3-scale max pyramid.
// Memory-bound gather (~30 MB total traffic -> ~1.3 us at 23.3 TB/s peak HBM;
// input is L2-resident on graph replay). No matmul content (max-reduction), so
// WMMA is inapplicable; we use the gfx1250 async global->LDS gather path
// (ASYNCcnt) in GVS addressing mode (SGPR64 base + per-lane VGPR32 byte offset)
// to stage the clamped patch tile into LDS and serve all three pooling scales
// from it.

#define IMG       224
#define PATCH_SZ  128
#define HALF      64
#define NTHREADS  256

__global__ __launch_bounds__(NTHREADS) void retina_kernel(
    const float* __restrict__ x,   // (B, 3, 224, 224)
    const float* __restrict__ l,   // (B, 2) in [-1, 1)
    float* __restrict__ out) {     // (B, 9, 32, 32)
  __shared__ float patch[PATCH_SZ * PATCH_SZ];  // 64 KB of the 320 KB/WGP LDS

  const int wg  = blockIdx.x;        // 0 .. B*3-1
  const int b   = wg / 3;
  const int c   = wg - b * 3;
  const int tid = threadIdx.x;

  // lidx = (int)(224 * (l + 1) / 2), then dynamic_slice start-clamp to [0, 352-128].
  const float l0 = l[b * 2 + 0];     // H start
  const float l1 = l[b * 2 + 1];     // W start
  int h0 = (int)(224.0f * (l0 + 1.0f) / 2.0f);
  int w0 = (int)(224.0f * (l1 + 1.0f) / 2.0f);
  h0 = min(max(h0, 0), IMG);
  w0 = min(max(w0, 0), IMG);

  // Wave-uniform channel base -> SGPR pair for GVS-mode async loads.
  const float* __restrict__ src = x + (size_t)(b * 3 + c) * (IMG * IMG);

  // ---- Stage edge-clamped 128x128 patch into LDS via async gather ----------
  // Thread layout: tid&127 -> column (clamp hoisted, loop-invariant),
  // row = (tid>>7) + 2k. Each unrolled iteration: one v_add (row), one
  // v_med3 (clamp), one mad (byte offset), one async gather. Global address
  // is SGPR base + 32-bit VGPR offset (GVS) -> no per-lane 64-bit adds.
  {
    const int cc = tid & (PATCH_SZ - 1);
    const int r0 = tid >> 7;                       // 0 or 1
    const int sw = min(max(w0 - HALF + cc, 0), IMG - 1);
    const unsigned lds_base = (unsigned)(size_t)patch;  // flat addr low 32 = LDS addr
#pragma unroll
    for (int k = 0; k < PATCH_SZ / 2; ++k) {
      const int r  = r0 + 2 * k;
      const int sh = min(max(h0 - HALF + r, 0), IMG - 1);
      const unsigned goff     = (unsigned)((sh * IMG + sw) * 4);   // byte offset
      const unsigned lds_addr = lds_base + (unsigned)((r * PATCH_SZ + cc) * 4);
      asm volatile("global_load_async_to_lds_b32 %0, %1, %2"
                   :: "v"(lds_addr), "v"(goff), "s"(src)
                   : "memory");
    }
  }
  asm volatile("s_wait_asynccnt 0x0" ::: "memory");  // my async writes landed in LDS
  __syncthreads();                                   // everyone's writes visible

  // ---- Three pooling scales from the LDS tile ------------------------------
  // out channel = scale*3 + c; each plane is 32x32.
  const size_t out_base = (size_t)b * 9 * 1024;
#pragma unroll
  for (int t = 0; t < 1024 / NTHREADS; ++t) {
    const int idx = tid + t * NTHREADS;        // 0..1023
    const int i = idx >> 5;
    const int j = idx & 31;

    // scale 0: 4x4 max over patch[4i..4i+3][4j..4j+3]
    {
      const float* p = &patch[(4 * i) * PATCH_SZ + 4 * j];
      float m = p[0];
#pragma unroll
      for (int rr = 0; rr < 4; ++rr)
#pragma unroll
        for (int cc2 = 0; cc2 < 4; ++cc2)
          m = fmaxf(m, p[rr * PATCH_SZ + cc2]);
      out[out_base + (size_t)(0 + c) * 1024 + idx] = m;
    }

    // scale 1: 2x2 max over central 64x64, i.e. patch[32+2i][32+2j] block
    {
      const float* p = &patch[(32 + 2 * i) * PATCH_SZ + (32 + 2 * j)];
      const float m = fmaxf(fmaxf(p[0], p[1]),
                            fmaxf(p[PATCH_SZ], p[PATCH_SZ + 1]));
      out[out_base + (size_t)(3 + c) * 1024 + idx] = m;
    }

    // scale 2: central 32x32 passthrough: patch[48+i][48+j]
    out[out_base + (size_t)(6 + c) * 1024 + idx] =
        patch[(48 + i) * PATCH_SZ + (48 + j)];
  }
}

extern "C" void kernel_launch(void* const* d_in, const int* in_sizes, int n_in,
                              void* d_out, int out_size, void* d_ws, size_t ws_size,
                              hipStream_t stream) {
  (void)n_in; (void)out_size; (void)d_ws; (void)ws_size;
  const float* x = (const float*)d_in[0];
  const float* l = (const float*)d_in[1];
  float* out = (float*)d_out;

  const int B = in_sizes[1] / 2;     // 128 per reference
  retina_kernel<<<dim3(B * 3), dim3(NTHREADS), 0, stream>>>(x, l, out);
}